// UnnaturalStates_45767171506789
// MI455X (gfx1250) — compile-verified
//
#include <hip/hip_runtime.h>

// ---------------------------------------------------------------------------
// Blocked linear scan for  x[t+1] = A22 x[t] + K2 (y[t]-y1[t]),
// y_out[t] = Cy2 x[t],  T=262144, N2=64, P=32, on gfx1250 (wave32, WMMA).
//
// Row-vector convention: s_{t+1} = s_t @ A22^T + u_t, so every update is a
// right-multiply that maps onto V_WMMA_F32_16X16X4_F32 with 16 independent
// chunks batched as the 16 rows of the A operand.
//
// 3-level scan, L=64 at each level (T = 64*64*64):
//   pass1   : 4096 chunks x 64 steps, 16 chunks/wave, 256 waves parallel
//   kpow    : P1=(A^T)^64, P2=P1^64 by repeated squaring (WMMA)
//   klevel2 : scan of the 4096 chunk summaries (WMMA + tiny VALU compose)
//   pass3   : recompute with correct start states, fused Cy2 GEMM, outputs
// ---------------------------------------------------------------------------

typedef __attribute__((ext_vector_type(2))) float v2f;
typedef __attribute__((ext_vector_type(8))) float v8f;

__device__ __forceinline__ v8f wmma4(v2f a, v2f b, v8f c) {
  // D(16x16,f32) = A(16x4,f32) * B(4x16,f32) + C
  return __builtin_amdgcn_wmma_f32_16x16x4_f32(false, a, false, b, (short)0, c,
                                               false, false);
}

constexpr int T   = 262144;
constexpr int PD  = 32;   // channels
constexpr int ND  = 64;   // state dim
constexpr int L   = 64;   // steps per chunk (all levels)
constexpr int C1  = T / L;          // 4096 level-1 chunks
constexpr int WPB = 4;              // waves per block (pass kernels)
constexpr int GPW = 16;             // chunks per wave (= WMMA M tile)

// Pack a KxN right-multiply matrix W into WMMA-B lane layout tiles:
// tile t = kb*NB+nb, entry = t*64 + lane*2 + w, value = W[kb*4+w+2*(l>>4)][nb*16+(l&15)].
// transpose=true means W(i,j) = src[j*K + i] (src is NxK row-major).
__device__ __forceinline__ void packB(float* dst, const float* src, int K,
                                      int N, bool transpose, int tid,
                                      int nthreads) {
  const int NB = N >> 4, KB = K >> 2;
  const int total = KB * NB * 64;
  for (int idx = tid; idx < total; idx += nthreads) {
    int t = idx >> 6, rem = idx & 63, l = rem >> 1, w = rem & 1;
    int kb = t / NB, nb = t % NB;
    int i = kb * 4 + w + 2 * (l >> 4);
    int j = nb * 16 + (l & 15);
    dst[idx] = transpose ? src[j * K + i] : src[i * N + j];
  }
}

// -------------------------- kpow: matrix powers ----------------------------
__global__ __launch_bounds__(32) void kpow(const float* __restrict__ A22,
                                           float* __restrict__ P1,
                                           float* __restrict__ P2) {
  __shared__ __align__(16) float M[4096];
  __shared__ __align__(16) float M2[4096];
  __shared__ __align__(16) float Bp[4096];
  const int lane = threadIdx.x;

  // M = A22^T  (W(i,j) = A22[j][i])
  for (int idx = lane; idx < 4096; idx += 32) {
    int i = idx >> 6, j = idx & 63;
    M[idx] = A22[j * 64 + i];
  }
  for (int sq = 0; sq < 12; sq++) {           // ^2 six times -> ^64; six more -> ^4096
    packB(Bp, M, 64, 64, false, lane, 32);    // same-wave LDS is in-order
    for (int m = 0; m < 4; m++) {
      v2f a[16];
#pragma unroll
      for (int kb = 0; kb < 16; kb++)
        a[kb] = *(const v2f*)(M + (m * 16 + (lane & 15)) * 64 + kb * 4 +
                              2 * (lane >> 4));
#pragma unroll
      for (int nb = 0; nb < 4; nb++) {
        v8f acc = {};
#pragma unroll
        for (int kb = 0; kb < 16; kb++)
          acc = wmma4(a[kb], *(const v2f*)(Bp + (kb * 4 + nb) * 64 + lane * 2),
                      acc);
#pragma unroll
        for (int r = 0; r < 8; r++)
          M2[(m * 16 + r + 8 * (lane >> 4)) * 64 + nb * 16 + (lane & 15)] =
              acc[r];
      }
    }
    for (int idx = lane; idx < 4096; idx += 32) M[idx] = M2[idx];
    if (sq == 5)
      for (int idx = lane; idx < 4096; idx += 32) P1[idx] = M[idx];
    if (sq == 11)
      for (int idx = lane; idx < 4096; idx += 32) P2[idx] = M[idx];
  }
}

// ------------------- pass1: chunk-local sums (from zero) -------------------
__global__ __launch_bounds__(128) void kpass1(const float* __restrict__ y,
                                              const float* __restrict__ y1,
                                              const float* __restrict__ A22,
                                              const float* __restrict__ K2,
                                              float* __restrict__ S1) {
  __shared__ __align__(16) float ATB[4096];            // A^T in B layout
  __shared__ __align__(16) float K2TB[2048];           // K2^T in B layout
  __shared__ __align__(16) float state[WPB][GPW * ND]; // 16x64 per wave
  const int tid = threadIdx.x, lane = tid & 31, wid = tid >> 5;
  packB(ATB, A22, 64, 64, true, tid, blockDim.x);
  packB(K2TB, K2, 32, 64, true, tid, blockDim.x);
  __syncthreads();

  float* st = &state[wid][0];
  const int base = (blockIdx.x * WPB + wid) * GPW;  // first chunk of group
  for (int idx = lane; idx < GPW * ND; idx += 32) st[idx] = 0.f;

  const int ca = 2 * (lane >> 4);  // K sub-offset for this lane half
  v8f zero8 = {};
#pragma unroll 1
  for (int j = 0; j < L; j++) {
    const int trow = (base + (lane & 15)) * L + j;
    // U = dY @ K2^T (fused input GEMM)
    v8f acc[4] = {zero8, zero8, zero8, zero8};
#pragma unroll
    for (int kb = 0; kb < 8; kb++) {
      int off = trow * PD + kb * 4 + ca;
      v2f a = *(const v2f*)(y + off);
      v2f b = *(const v2f*)(y1 + off);
      v2f dy = {a[0] - b[0], a[1] - b[1]};
#pragma unroll
      for (int nb = 0; nb < 4; nb++)
        acc[nb] = wmma4(dy, *(const v2f*)(K2TB + (kb * 4 + nb) * 64 + lane * 2),
                        acc[nb]);
    }
    // S = S @ A^T + U
    v2f s[16];
#pragma unroll
    for (int kb = 0; kb < 16; kb++)
      s[kb] = *(const v2f*)(st + (lane & 15) * 64 + kb * 4 + ca);
#pragma unroll
    for (int kb = 0; kb < 16; kb++)
#pragma unroll
      for (int nb = 0; nb < 4; nb++)
        acc[nb] = wmma4(s[kb], *(const v2f*)(ATB + (kb * 4 + nb) * 64 + lane * 2),
                        acc[nb]);
#pragma unroll
    for (int nb = 0; nb < 4; nb++)
#pragma unroll
      for (int r = 0; r < 8; r++)
        st[(r + 8 * (lane >> 4)) * 64 + nb * 16 + (lane & 15)] = acc[nb][r];
  }
  // state rows = chunk-local summaries s1[c]
  for (int idx = lane; idx < GPW * ND; idx += 32) {
    int m = idx >> 6, col = idx & 63;
    S1[(base + m) * 64 + col] = st[idx];
  }
}

// ------------- klevel2: scan the 4096 summaries -> start states ------------
__global__ __launch_bounds__(128) void klevel2(const float* __restrict__ S1,
                                               const float* __restrict__ P1,
                                               const float* __restrict__ P2,
                                               float* __restrict__ W0g,
                                               float* __restrict__ X0) {
  __shared__ __align__(16) float P1B[4096];
  __shared__ __align__(16) float state[4][GPW * ND];
  __shared__ __align__(16) float P2L[4096];
  __shared__ __align__(16) float wbuf[64];
  const int tid = threadIdx.x, lane = tid & 31, wid = tid >> 5;
  const int ca = 2 * (lane >> 4);
  v8f zero8 = {};

  // ---- phase A: 64 super-chunks (16 per wave), local sums with P1 ----
  packB(P1B, P1, 64, 64, false, tid, blockDim.x);
  for (int idx = tid; idx < 4 * GPW * ND; idx += blockDim.x)
    state[idx >> 10][idx & 1023] = 0.f;
  __syncthreads();

  float* st = &state[wid][0];
#pragma unroll 1
  for (int j = 0; j < L; j++) {
    v8f acc[4];
#pragma unroll
    for (int nb = 0; nb < 4; nb++) {
      acc[nb] = zero8;
#pragma unroll
      for (int r = 0; r < 8; r++) {
        int c = (wid * 16 + r + 8 * (lane >> 4)) * 64 + j;
        acc[nb][r] = S1[c * 64 + nb * 16 + (lane & 15)];
      }
    }
    v2f s[16];
#pragma unroll
    for (int kb = 0; kb < 16; kb++)
      s[kb] = *(const v2f*)(st + (lane & 15) * 64 + kb * 4 + ca);
#pragma unroll
    for (int kb = 0; kb < 16; kb++)
#pragma unroll
      for (int nb = 0; nb < 4; nb++)
        acc[nb] = wmma4(s[kb], *(const v2f*)(P1B + (kb * 4 + nb) * 64 + lane * 2),
                        acc[nb]);
#pragma unroll
    for (int nb = 0; nb < 4; nb++)
#pragma unroll
      for (int r = 0; r < 8; r++)
        st[(r + 8 * (lane >> 4)) * 64 + nb * 16 + (lane & 15)] = acc[nb][r];
  }
  __syncthreads();

  // ---- phase B: 64-step compose with P2 (tiny, VALU) ----
  for (int idx = tid; idx < 4096; idx += blockDim.x) P2L[idx] = P2[idx];
  __syncthreads();
  float wcur = 0.f;
  for (int c2 = 0; c2 < 64; c2++) {
    if (tid < 64) {
      W0g[c2 * 64 + tid] = wcur;   // E[c2*64]
      wbuf[tid] = wcur;
    }
    __syncthreads();
    float wnext = 0.f;
    if (tid < 64) {
      float a2 = state[c2 >> 4][(c2 & 15) * 64 + tid];  // s2[c2]
      for (int i = 0; i < 64; i++) a2 += wbuf[i] * P2L[i * 64 + tid];
      wnext = a2;
    }
    __syncthreads();
    wcur = wnext;
  }

  // ---- phase C: recompute level-2 -> X0[c] for all 4096 chunks ----
  for (int idx = tid; idx < 4096; idx += blockDim.x) {
    int c2 = idx >> 6, col = idx & 63;
    state[c2 >> 4][(c2 & 15) * 64 + col] = W0g[idx];
  }
  __syncthreads();
#pragma unroll 1
  for (int j = 0; j < L; j++) {
    v2f s[16];
#pragma unroll
    for (int kb = 0; kb < 16; kb++)
      s[kb] = *(const v2f*)(st + (lane & 15) * 64 + kb * 4 + ca);
    // X0[c] = current state (row-vector) for c = (wid*16 + row)*64 + j
    {
      int c = (wid * 16 + (lane & 15)) * 64 + j;
#pragma unroll
      for (int kb = 0; kb < 16; kb++)
        *(v2f*)(X0 + (size_t)c * 64 + kb * 4 + ca) = s[kb];
    }
    v8f acc[4];
#pragma unroll
    for (int nb = 0; nb < 4; nb++) {
      acc[nb] = zero8;
#pragma unroll
      for (int r = 0; r < 8; r++) {
        int c = (wid * 16 + r + 8 * (lane >> 4)) * 64 + j;
        acc[nb][r] = S1[c * 64 + nb * 16 + (lane & 15)];
      }
    }
#pragma unroll
    for (int kb = 0; kb < 16; kb++)
#pragma unroll
      for (int nb = 0; nb < 4; nb++)
        acc[nb] = wmma4(s[kb], *(const v2f*)(P1B + (kb * 4 + nb) * 64 + lane * 2),
                        acc[nb]);
#pragma unroll
    for (int nb = 0; nb < 4; nb++)
#pragma unroll
      for (int r = 0; r < 8; r++)
        st[(r + 8 * (lane >> 4)) * 64 + nb * 16 + (lane & 15)] = acc[nb][r];
  }
}

// --------- pass3: recompute chunks from X0, fused Cy2 GEMM, outputs --------
__global__ __launch_bounds__(128) void kpass3(
    const float* __restrict__ y, const float* __restrict__ y1,
    const float* __restrict__ A22, const float* __restrict__ K2,
    const float* __restrict__ Cy2, const float* __restrict__ X0,
    float* __restrict__ x2o, float* __restrict__ yo) {
  __shared__ __align__(16) float ATB[4096];
  __shared__ __align__(16) float K2TB[2048];
  __shared__ __align__(16) float CyTB[2048];
  __shared__ __align__(16) float state[WPB][GPW * ND];
  const int tid = threadIdx.x, lane = tid & 31, wid = tid >> 5;
  packB(ATB, A22, 64, 64, true, tid, blockDim.x);
  packB(K2TB, K2, 32, 64, true, tid, blockDim.x);
  packB(CyTB, Cy2, 64, 32, true, tid, blockDim.x);
  __syncthreads();

  float* st = &state[wid][0];
  const int base = (blockIdx.x * WPB + wid) * GPW;
  for (int idx = lane; idx < GPW * ND; idx += 32) {
    int m = idx >> 6, col = idx & 63;
    st[idx] = X0[(size_t)(base + m) * 64 + col];  // x2 at chunk start
  }
  const int ca = 2 * (lane >> 4);
  v8f zero8 = {};
#pragma unroll 1
  for (int j = 0; j < L; j++) {
    const int trow = (base + (lane & 15)) * L + j;  // global t for this lane's row
    v2f s[16];
#pragma unroll
    for (int kb = 0; kb < 16; kb++)
      s[kb] = *(const v2f*)(st + (lane & 15) * 64 + kb * 4 + ca);
    // x2[t] (state BEFORE update)
#pragma unroll
    for (int kb = 0; kb < 16; kb++)
      *(v2f*)(x2o + (size_t)trow * 64 + kb * 4 + ca) = s[kb];
    // y_out[t] = x2[t] @ Cy2^T
    v8f yacc[2] = {zero8, zero8};
#pragma unroll
    for (int kb = 0; kb < 16; kb++)
#pragma unroll
      for (int nb = 0; nb < 2; nb++)
        yacc[nb] = wmma4(s[kb], *(const v2f*)(CyTB + (kb * 2 + nb) * 64 + lane * 2),
                         yacc[nb]);
#pragma unroll
    for (int nb = 0; nb < 2; nb++)
#pragma unroll
      for (int r = 0; r < 8; r++) {
        int tr = (base + r + 8 * (lane >> 4)) * L + j;
        float v = yacc[nb][r];
        if (tr == T - 1) v = 0.f;  // reference zeroes the last row
        yo[(size_t)tr * PD + nb * 16 + (lane & 15)] = v;
      }
    // U = dY @ K2^T, then S = S @ A^T + U
    v8f acc[4] = {zero8, zero8, zero8, zero8};
#pragma unroll
    for (int kb = 0; kb < 8; kb++) {
      int off = trow * PD + kb * 4 + ca;
      v2f a = *(const v2f*)(y + off);
      v2f b = *(const v2f*)(y1 + off);
      v2f dy = {a[0] - b[0], a[1] - b[1]};
#pragma unroll
      for (int nb = 0; nb < 4; nb++)
        acc[nb] = wmma4(dy, *(const v2f*)(K2TB + (kb * 4 + nb) * 64 + lane * 2),
                        acc[nb]);
    }
#pragma unroll
    for (int kb = 0; kb < 16; kb++)
#pragma unroll
      for (int nb = 0; nb < 4; nb++)
        acc[nb] = wmma4(s[kb], *(const v2f*)(ATB + (kb * 4 + nb) * 64 + lane * 2),
                        acc[nb]);
#pragma unroll
    for (int nb = 0; nb < 4; nb++)
#pragma unroll
      for (int r = 0; r < 8; r++)
        st[(r + 8 * (lane >> 4)) * 64 + nb * 16 + (lane & 15)] = acc[nb][r];
  }
}

extern "C" void kernel_launch(void* const* d_in, const int* in_sizes, int n_in,
                              void* d_out, int out_size, void* d_ws,
                              size_t ws_size, hipStream_t stream) {
  const float* y   = (const float*)d_in[0];
  const float* y1  = (const float*)d_in[1];
  const float* A22 = (const float*)d_in[2];
  const float* K2  = (const float*)d_in[3];
  const float* Cy2 = (const float*)d_in[4];
  float* out = (float*)d_out;
  float* x2o = out;                       // (T, 64)
  float* yo  = out + (size_t)T * ND;      // (T, 32)

  float* ws = (float*)d_ws;               // needs ~2.1 MB
  float* P1 = ws;                         // 4096
  float* P2 = P1 + 4096;                  // 4096
  float* S1 = P2 + 4096;                  // C1*64 = 262144
  float* X0 = S1 + (size_t)C1 * ND;       // 262144
  float* W0 = X0 + (size_t)C1 * ND;       // 4096

  kpass1<<<C1 / (WPB * GPW), WPB * 32, 0, stream>>>(y, y1, A22, K2, S1);
  kpow<<<1, 32, 0, stream>>>(A22, P1, P2);
  klevel2<<<1, 128, 0, stream>>>(S1, P1, P2, W0, X0);
  kpass3<<<C1 / (WPB * GPW), WPB * 32, 0, stream>>>(y, y1, A22, K2, Cy2, X0,
                                                    x2o, yo);
}